// SpatialTransformer_274877907312
// MI455X (gfx1250) — compile-verified
//
#include <hip/hip_runtime.h>
#include <stdint.h>

// Problem shape (fixed by reference setup_inputs)
#define BB   16
#define CC   4
#define HH   768
#define WW   768
#define TILE 256              // one block = one 256-pixel row segment

typedef unsigned int u32x4 __attribute__((ext_vector_type(4)));
typedef int          i32x8 __attribute__((ext_vector_type(8)));
typedef int          i32x4 __attribute__((ext_vector_type(4)));

__global__ __launch_bounds__(TILE) void spatial_warp2d_kernel(
    const float* __restrict__ img,   // [B,C,H,W]
    const float* __restrict__ trf,   // [B,2,H,W]  (dy plane, dx plane)
    float* __restrict__ out)         // [B,C,H,W]
{
    __shared__ float lds_trf[2 * TILE];   // row0 = dy, row1 = dx (TDM writes both)

    const int HW = HH * WW;
    // 3D grid: no div/mod needed to decode the block position.
    int x0 = (int)blockIdx.x * TILE;      // tile start within row
    int y  = (int)blockIdx.y;             // image row
    int b  = (int)blockIdx.z;             // batch

    const float* trf_tile = trf + (size_t)b * 2 * HW + (size_t)y * WW + x0;

#if __has_builtin(__builtin_amdgcn_tensor_load_to_lds) && __has_builtin(__builtin_amdgcn_s_wait_tensorcnt)
    // ---- Tensor Data Mover: DMA the 2x256 f32 displacement tile into LDS ----
    // D# per CDNA5 ISA: 2D tile, dim0 stride = H*W elements
    // jumps from the dy plane row to the dx plane row of the same (y, x0..x0+255).
    if (threadIdx.x < 32u) {              // one wave issues the DMA (TDM ignores EXEC)
        uint64_t ga      = (uint64_t)(uintptr_t)trf_tile;
        unsigned lds_off = (unsigned)(uintptr_t)(void*)&lds_trf[0]; // LDS byte offset (low 32 bits of flat addr)

        u32x4 g0;
        g0[0] = 1u;                                   // count=1 (valid), user-mode, no gather
        g0[1] = lds_off;                              // lds_addr (bytes)
        g0[2] = (unsigned)(ga & 0xFFFFFFFFull);       // global_addr[31:0]
        g0[3] = (unsigned)((ga >> 32) & 0x1FFFFFFull) // global_addr[56:32]
              | (2u << 30);                           // type = 2 ("image")

        i32x8 g1;
        g1[0] = (int)(2u << 16);                      // workgroup_mask=0 | data_size=2 (4 bytes)
        g1[1] = (int)((unsigned)TILE << 16);          // tensor_dim0[15:0] = 256
        g1[2] = (int)(2u << 16);                      // tensor_dim0[31:16]=0 | tensor_dim1[15:0]=2
        g1[3] = (int)((unsigned)TILE << 16);          // tensor_dim1 hi=0 | tile_dim0=256
        g1[4] = 2;                                    // tile_dim1=2, tile_dim2=0
        g1[5] = HW;                                   // tensor_dim0_stride[31:0] = H*W elements
        g1[6] = 0;                                    // stride hi / dim1_stride lo
        g1[7] = 0;                                    // dim1_stride hi (unused, 2D tile)

        i32x4 gz4 = {0, 0, 0, 0};                     // groups 2/3 unused for 2D tensor
        i32x8 gz8 = {0, 0, 0, 0, 0, 0, 0, 0};         // extra group (clang-23 6-arg form)

        __builtin_amdgcn_tensor_load_to_lds(g0, g1, gz4, gz4, gz8, 0 /*cpol*/);
        __builtin_amdgcn_s_wait_tensorcnt(0);         // s_wait_tensorcnt 0
    }
#else
    // Fallback staging path: cooperative coalesced LDS fill
    lds_trf[threadIdx.x]        = trf_tile[threadIdx.x];
    lds_trf[TILE + threadIdx.x] = trf_tile[HW + threadIdx.x];
#endif
    __syncthreads();

    int   x  = x0 + (int)threadIdx.x;
    float dy = lds_trf[threadIdx.x];
    float dx = lds_trf[TILE + threadIdx.x];

    // Absolute sample location, edge-clamped (fill_value=None semantics)
    const float my = (float)(HH - 1);
    const float mx = (float)(WW - 1);
    float ly  = (float)y + dy;
    float lx  = (float)x + dx;
    float cy  = fminf(fmaxf(ly, 0.0f), my);
    float cx  = fminf(fmaxf(lx, 0.0f), mx);
    float y0f = fminf(fmaxf(floorf(ly), 0.0f), my);
    float x0f = fminf(fmaxf(floorf(lx), 0.0f), mx);
    float y1f = fminf(y0f + 1.0f, my);
    float x1f = fminf(x0f + 1.0f, mx);

    // neurite convention: floor corner weight d1 = loc1 - clipped; ceil = 1 - d1
    float wy0 = y1f - cy;
    float wx0 = x1f - cx;
    float wy1 = 1.0f - wy0;
    float wx1 = 1.0f - wx0;

    int iy0 = (int)y0f, iy1 = (int)y1f;
    int ix0 = (int)x0f, ix1 = (int)x1f;

    int o00 = iy0 * WW + ix0;
    int o01 = iy0 * WW + ix1;
    int o10 = iy1 * WW + ix0;
    int o11 = iy1 * WW + ix1;

    float w00 = wy0 * wx0;
    float w01 = wy0 * wx1;
    float w10 = wy1 * wx0;
    float w11 = wy1 * wx1;

    size_t base = (size_t)b * CC * HW;
    size_t opix = (size_t)y * WW + (size_t)x;

#pragma unroll
    for (int c = 0; c < CC; ++c) {
        const float* p = img + base + (size_t)c * HW;
        float v = w00 * p[o00] + w01 * p[o01] + w10 * p[o10] + w11 * p[o11];
        // Output is never re-read: bypass-cache store so L2 keeps img rows hot.
        __builtin_nontemporal_store(v, out + base + (size_t)c * HW + opix);
    }
}

extern "C" void kernel_launch(void* const* d_in, const int* in_sizes, int n_in,
                              void* d_out, int out_size, void* d_ws, size_t ws_size,
                              hipStream_t stream) {
    const float* img = (const float*)d_in[0];  // [16,4,768,768] f32
    const float* trf = (const float*)d_in[1];  // [16,2,768,768] f32
    float*       out = (float*)d_out;          // [16,4,768,768] f32

    dim3 grid(WW / TILE, HH, BB);              // 3 x 768 x 16 = 36864 blocks
    dim3 block(TILE);                          // 8 waves of 32
    spatial_warp2d_kernel<<<grid, block, 0, stream>>>(img, trf, out);
}